// MovingNormalizationLayer_62105227100226
// MI455X (gfx1250) — compile-verified
//
#include <hip/hip_runtime.h>
#include <cfloat>
#include <cstdint>

// Moving normalization over depth (last dim), window=100 (pad 50 left / 49 right,
// fixed n=100), x: 32768 rows x 1024 fp32. Memory-bound: 268 MB @ 23.3 TB/s ~ 11.5 us.
//
// Layout: 1 wave32 per row, 8 waves / block. Row staged in LDS with CDNA5 async
// copies (global_load_async_to_lds_b32, ASYNCcnt). LDS rows are padded 1 word per
// 32 (P(i) = i + (i>>5)) so every compute access has lane-stride 33 words ->
// conflict-free across all 64 banks. Init window sums come from per-chunk partial
// sums + wave shuffles; the window then slides in O(1) per output. Lane L's 32
// outputs are one 128B cacheline -> direct float4 stores fill lines in L2.

#define DEPTH 1024
#define RPB   8                      // rows (waves) per block
#define ROWW  (DEPTH + DEPTH / 32)   // 1056 padded words per LDS row

__global__ __launch_bounds__(256)
void moving_norm_depth100_kernel(const float* __restrict__ x,
                                 float* __restrict__ out,
                                 int nrows) {
    __shared__ __align__(16) float s_in[RPB * ROWW];

    const int lane = threadIdx.x & 31;
    const int wave = threadIdx.x >> 5;
    const int row  = blockIdx.x * RPB + wave;
    if (row >= nrows) return;                      // wave-uniform

    const float* __restrict__ grow = x   + (size_t)row * DEPTH;
    float*       __restrict__ orow = out + (size_t)row * DEPTH;
    float* srow = &s_in[wave * ROWW];

    // Low 32 bits of the flat shared-aperture address = LDS byte offset
    // relative to LDS_BASE (what the async op's VDST operand addresses).
    const unsigned lds_row = (unsigned)(uintptr_t)srow;

    // --- CDNA5 async global->LDS fill (padded layout) ---------------------
    // word w = it*32 + lane ; padded word P(w) = 33*it + lane.
    // Per instruction: 128 B contiguous global read, lane-stride-1 LDS write.
    #pragma unroll
    for (int it = 0; it < 32; ++it) {
        const unsigned laddr = lds_row + 132u * it + 4u * lane;
        const float*   gaddr = grow + 32 * it + lane;
        asm volatile("global_load_async_to_lds_b32 %0, %1, off"
                     :: "v"(laddr), "v"(gaddr) : "memory");
    }
    asm volatile("s_wait_asynccnt 0" ::: "memory");
    // Producer/consumer is intra-wave only -> no workgroup barrier needed.

    // --- per-chunk partial sums (own 32 values; conflict-free LDS reads) --
    // Own chunk d0 = 32*lane starts at padded word 33*lane.
    const float* srowL = srow + 33 * lane;
    float s1t = 0.f, s2t = 0.f;   // full chunk
    float h1  = 0.f, h2  = 0.f;   // head 18  (m = 0..17)
    float t1  = 0.f, t2  = 0.f;   // tail 18  (m = 14..31)
    #pragma unroll
    for (int m = 0; m < 32; ++m) {
        const float v  = srowL[m];
        const float vv = v * v;
        s1t += v;  s2t += vv;
        if (m < 18)  { h1 += v; h2 += vv; }
        if (m >= 14) { t1 += v; t2 += vv; }
    }

    // --- init window sums for d0: [d0-50, d0+49] --------------------------
    //   = tail18(chunk L-2) + full(chunk L-1) + full(own) + head18(chunk L+1)
    float s1 = s1t, s2 = s2t;
    {
        const float a = __shfl_up(s1t, 1, 32), b = __shfl_up(s2t, 1, 32);
        if (lane >= 1) { s1 += a; s2 += b; }
    }
    {
        const float a = __shfl_up(t1, 2, 32),  b = __shfl_up(t2, 2, 32);
        if (lane >= 2) { s1 += a; s2 += b; }
    }
    {
        const float a = __shfl_down(h1, 1, 32), b = __shfl_down(h2, 1, 32);
        if (lane <= 30) { s1 += a; s2 += b; }
    }

    // --- slide over the lane's 32 outputs ---------------------------------
    // Padded offsets relative to srowL (constant-folded by full unroll):
    //   x[d]    -> srowL[k]
    //   x[d+50] -> srowL[k + 51 + (k>=14)]
    //   x[d-50] -> srowL[k - 52 + (k>=18)]
    float rq[4];
    #pragma unroll
    for (int k = 0; k < 32; ++k) {
        const int   d     = lane * 32 + k;
        const float xv    = srowL[k];
        const float mean  = s1 * 0.01f;
        const float mean2 = s2 * 0.01f;
        const float var   = fmaf(-mean, mean, mean2);  // may be tiny-negative
        const float rs    = rsqrtf(var);               // NaN if var<0, +inf if var==0
        float r = (xv - mean) * rs;
        // jnp.nan_to_num(nan=0, posinf=+FLT_MAX, neginf=-FLT_MAX)
        r = (r != r) ? 0.0f : fminf(fmaxf(r, -FLT_MAX), FLT_MAX);
        rq[k & 3] = r;
        if ((k & 3) == 3) {
            const float4 v4 = make_float4(rq[0], rq[1], rq[2], rq[3]);
            ((float4*)orow)[lane * 8 + (k >> 2)] = v4;   // lane-private 128B line
        }

        // window [d-50, d+49] -> [d-49, d+50]
        const int   ja   = d + 50;
        const int   jr   = d - 50;
        const int   offp = k + 51 + (k >= 14 ? 1 : 0);
        const int   offm = k - 52 + (k >= 18 ? 1 : 0);
        const float va   = (ja < DEPTH) ? srowL[offp] : 0.0f;
        const float vr   = (jr >= 0)    ? srowL[offm] : 0.0f;
        s1 += va - vr;
        s2 += fmaf(va, va, -(vr * vr));
    }
}

extern "C" void kernel_launch(void* const* d_in, const int* in_sizes, int n_in,
                              void* d_out, int out_size, void* d_ws, size_t ws_size,
                              hipStream_t stream) {
    const float* x  = (const float*)d_in[0];
    float* out      = (float*)d_out;
    const int total = in_sizes[0];          // 1*1*128*256*1024
    const int nrows = total / DEPTH;        // 32768
    const dim3 grid((nrows + RPB - 1) / RPB);
    moving_norm_depth100_kernel<<<grid, 256, 0, stream>>>(x, out, nrows);
}